// SSIM_90537910600252
// MI455X (gfx1250) — compile-verified
//
#include <hip/hip_runtime.h>

// SSIM 3x3 (reflect pad) for NCHW (32,3,384,640) fp32.
// Vertical 3-row box sums are done on the matrix pipe via
// V_WMMA_F32_16X16X4_F32 with a 0/1 band matrix, overlapping the
// pointwise SSIM math on the VALU. Memory-bound target: ~283 MB of
// mandatory HBM traffic @ 23.3 TB/s.

typedef float v2f __attribute__((ext_vector_type(2)));
typedef float v8f __attribute__((ext_vector_type(8)));

#define IMG_H 384
#define IMG_W 640
#define TILE_H 16
#define TILE_W 128
#define PAD_H 18            // TILE_H + 2 halo rows
#define PAD_W 130           // TILE_W + 2 halo cols
#define NQ 5                // Sx, Sy, Sxx, Syy, Sxy

__global__ __launch_bounds__(256) void ssim_wmma_kernel(
    const float* __restrict__ X, const float* __restrict__ Y,
    float* __restrict__ O)
{
    __shared__ float xs[PAD_H * PAD_W];
    __shared__ float ys[PAD_H * PAD_W];
    __shared__ float Hb[8][NQ][PAD_H][16];   // per-wave horizontal 3-sums

    const int tid  = threadIdx.x;
    const int lane = tid & 31;
    const int wv   = tid >> 5;               // wave id 0..7
    const int c0   = blockIdx.x * TILE_W;
    const int r0   = blockIdx.y * TILE_H;
    const long plane = blockIdx.z;

    const float* __restrict__ xp = X + plane * (long)(IMG_H * IMG_W);
    const float* __restrict__ yp = Y + plane * (long)(IMG_H * IMG_W);

    // ---- Phase 1: cooperative haloed tile load with reflect padding ----
    for (int t = tid; t < PAD_H * PAD_W; t += 256) {
        int i = t / PAD_W;
        int j = t - i * PAD_W;
        int rr = r0 - 1 + i;
        rr = rr < 0 ? -rr : (rr >= IMG_H ? 2 * IMG_H - 2 - rr : rr);
        int cc = c0 - 1 + j;
        cc = cc < 0 ? -cc : (cc >= IMG_W ? 2 * IMG_W - 2 - cc : cc);
        long g = (long)rr * IMG_W + cc;
        xs[t] = xp[g];
        ys[t] = yp[g];
    }
    __syncthreads();

    // ---- Phase 2: per-wave horizontal 3-sums for 18 rows x 16 cols ----
    // Output col n of this wave <-> LDS col (16*wv + n) as left-halo index.
    const int colbase = wv * 16;
    for (int p = lane; p < PAD_H * 16; p += 32) {   // 288/32 = 9 uniform iters
        int r = p >> 4;
        int n = p & 15;
        const float* xr = &xs[r * PAD_W + colbase + n];
        const float* yr = &ys[r * PAD_W + colbase + n];
        float x0 = xr[0], x1 = xr[1], x2 = xr[2];
        float y0 = yr[0], y1 = yr[1], y2 = yr[2];
        Hb[wv][0][r][n] = x0 + x1 + x2;
        Hb[wv][1][r][n] = y0 + y1 + y2;
        Hb[wv][2][r][n] = x0 * x0 + x1 * x1 + x2 * x2;
        Hb[wv][3][r][n] = y0 * y0 + y1 * y1 + y2 * y2;
        Hb[wv][4][r][n] = x0 * y0 + x1 * y1 + x2 * y2;
    }
    // Same-wave LDS producer->consumer: in-order, compiler inserts DS waits.

    // ---- Phase 3: vertical 3-row sums as band-matrix WMMA ----
    // D(16x16) = T(16x20) * H(20x16), T[m][k] = (m <= k <= m+2), rows 18/19 zero.
    // A layout (ISA 7.12.2, 32-bit A 16x4): lanes 0-15 M=0..15 hold K=k0+0/k0+1,
    // lanes 16-31 hold K=k0+2/k0+3. B assumed symmetric (rows k0+0/k0+1 in
    // lanes 0-15, k0+2/k0+3 in lanes 16-31).
    v8f acc[NQ];
    const v8f vzero = {0.f, 0.f, 0.f, 0.f, 0.f, 0.f, 0.f, 0.f};
#pragma unroll
    for (int q = 0; q < NQ; ++q) acc[q] = vzero;

    const int m  = lane & 15;    // A row (M) and B column (N) for this lane
    const int hi = lane >> 4;    // lane half

#pragma unroll
    for (int k0 = 0; k0 < 20; k0 += 4) {
        const int kb = k0 + hi * 2;
        v2f a;
        a.x = (kb     >= m && kb     <= m + 2) ? 1.0f : 0.0f;
        a.y = (kb + 1 >= m && kb + 1 <= m + 2) ? 1.0f : 0.0f;
#pragma unroll
        for (int q = 0; q < NQ; ++q) {
            int kc0 = kb     < PAD_H ? kb     : 0;
            int kc1 = kb + 1 < PAD_H ? kb + 1 : 0;
            float b0 = Hb[wv][q][kc0][m];
            float b1 = Hb[wv][q][kc1][m];
            if (kb     >= PAD_H) b0 = 0.0f;   // zero-pad K=18,19
            if (kb + 1 >= PAD_H) b1 = 0.0f;
            v2f b; b.x = b0; b.y = b1;
            acc[q] = __builtin_amdgcn_wmma_f32_16x16x4_f32(
                false, a, false, b, (short)0, acc[q], false, false);
        }
    }

    // ---- Phase 4: pointwise SSIM + store ----
    // C/D layout: VGPR j holds M=j (lanes 0-15) and M=j+8 (lanes 16-31).
    const float inv9 = 1.0f / 9.0f;
    const float C1 = 1e-4f, C2 = 9e-4f;
    float* __restrict__ op = O + plane * (long)(IMG_H * IMG_W);
    const int gcol = c0 + colbase + m;
#pragma unroll
    for (int j = 0; j < 8; ++j) {
        const int M = j + (hi << 3);
        float sx  = acc[0][j], sy  = acc[1][j];
        float sxx = acc[2][j], syy = acc[3][j], sxy = acc[4][j];
        float mux = sx * inv9, muy = sy * inv9;
        float varx = sxx * inv9 - mux * mux;
        float vary = syy * inv9 - muy * muy;
        float cov  = sxy * inv9 - mux * muy;
        float num = (2.0f * mux * muy + C1) * (2.0f * cov + C2);
        float den = (mux * mux + muy * muy + C1) * (varx + vary + C2);
        float s = num / den;
        s = s < 0.0f ? 0.0f : (s > 1.0f ? 1.0f : s);
        op[(long)(r0 + M) * IMG_W + gcol] = s;
    }
}

extern "C" void kernel_launch(void* const* d_in, const int* in_sizes, int n_in,
                              void* d_out, int out_size, void* d_ws, size_t ws_size,
                              hipStream_t stream) {
    const float* x = (const float*)d_in[0];
    const float* y = (const float*)d_in[1];
    float* out = (float*)d_out;
    const int planes = in_sizes[0] / (IMG_H * IMG_W);   // 32*3 = 96
    dim3 grid(IMG_W / TILE_W, IMG_H / TILE_H, planes);  // (5, 24, 96)
    ssim_wmma_kernel<<<grid, 256, 0, stream>>>(x, y, out);
}